// PINNLISTALNNModel_79010218377644
// MI455X (gfx1250) — compile-verified
//
#include <hip/hip_runtime.h>
#include <math.h>

// ---------------------------------------------------------------------------
// Fused PINN-LISTA-LNN forward for MI455X (gfx1250, wave32, WMMA).
//   Phase 1: 100-step liquid recurrence  h = clip(h*decay + DT*tanh(LN(h@W_rec + x_t*w_in + b_in)))
//   Phase 2: x = h @ Wx0^T + bx0
//   Phase 3: 8 ISTA iterations  x = softthr(y@We[k]^T + be[k] + x@Wr[k]^T)
//   Phase 4: power = x @ head_W^T + head_b ; outputs (power, x)
// All GEMMs use V_WMMA_F32_16X16X4_F32 (fp32, matches reference precision).
// tanh uses gfx1250's V_TANH_F32 TRANS op when the builtin is available.
// y staging uses GLOBAL_LOAD_ASYNC_TO_LDS_B32 (ASYNCcnt) when available.
// ---------------------------------------------------------------------------

typedef float v2f __attribute__((ext_vector_type(2)));
typedef float v8f __attribute__((ext_vector_type(8)));

typedef __attribute__((address_space(1))) int  gint_t;   // global (AS1)
typedef __attribute__((address_space(3))) int  sint_t;   // LDS (AS3)

#define NB     32768
#define WIN    100
#define HH     64
#define AD     64
#define KIT    8
#define NAPPL  4
#define DTC    0.1f
#define LNEPS  1e-5f

#define RPW 16              // batch rows per wave (WMMA M)
#define WPB 4               // waves per block
#define RPB (RPW * WPB)     // 64 rows per block
#define HPAD (HH + 2)       // LDS row padding (bank-conflict avoidance, keeps 8B align)
#define YPAD (RPB + 2)

#define WMMA_F32(a, b, c) \
  __builtin_amdgcn_wmma_f32_16x16x4_f32(false, (a), false, (b), (short)0, (c), false, false)

#if __has_builtin(__builtin_amdgcn_global_load_async_to_lds_b32)
#define USE_ASYNC_LDS 1
#endif

__device__ __forceinline__ float softplusf(float x) {
  return (x > 20.f) ? x : log1pf(expf(x));
}

__device__ __forceinline__ float fast_tanh(float x) {
#if __has_builtin(__builtin_amdgcn_tanhf)
  return __builtin_amdgcn_tanhf(x);       // V_TANH_F32 (TRANS32, co-executes with XDL)
#elif __has_builtin(__builtin_amdgcn_tanh_f32)
  return __builtin_amdgcn_tanh_f32(x);
#else
  return tanhf(x);
#endif
}

__device__ __forceinline__ void wait_async0() {
#if __has_builtin(__builtin_amdgcn_s_wait_asynccnt)
  __builtin_amdgcn_s_wait_asynccnt(0);
#else
  asm volatile("s_wait_asynccnt 0x0" ::: "memory");
#endif
}

__global__ __launch_bounds__(128, 1)
void pinn_lista_fused(const float* __restrict__ y,
                      const float* __restrict__ w_in,  const float* __restrict__ b_in,
                      const float* __restrict__ tau_p, const float* __restrict__ W_rec,
                      const float* __restrict__ ln_w,  const float* __restrict__ ln_b,
                      const float* __restrict__ Wx0,   const float* __restrict__ bx0,
                      const float* __restrict__ We,    const float* __restrict__ be,
                      const float* __restrict__ Wr,    const float* __restrict__ thr,
                      const float* __restrict__ head_W, const float* __restrict__ head_b,
                      float* __restrict__ out_power,   float* __restrict__ out_x)
{
  __shared__ float yT[WIN][YPAD];     // y transposed: yT[t][local_row]
  __shared__ float hbuf[RPB][HPAD];   // h (then x) in row-major [row][feature]

  const int tid  = threadIdx.x;
  const int lane = tid & 31;
  const int wave = tid >> 5;
  const int l16  = lane & 15;         // WMMA: A-row index / B,C,D column index
  const int hi   = lane >> 4;         // lane half selects K pair / M-half
  const int lrow = wave * RPW;        // this wave's local row base
  const size_t grow0 = (size_t)blockIdx.x * RPB;

  // ---- stage y into LDS (transposed) via async-to-LDS DMA when available ----
#ifdef USE_ASYNC_LDS
  for (int i = tid; i < RPB * WIN; i += blockDim.x) {
    int r = i / WIN, t = i - r * WIN;
    const float* g = &y[(grow0 + (size_t)r) * WIN + t];
    __builtin_amdgcn_global_load_async_to_lds_b32(
        (gint_t*)g, (sint_t*)&yT[t][r], 0, 0);
  }
  wait_async0();
#else
  for (int i = tid; i < RPB * WIN; i += blockDim.x) {
    int r = i / WIN, t = i - r * WIN;
    yT[t][r] = y[(grow0 + (size_t)r) * WIN + t];
  }
#endif

  // zero this wave's h slice
  for (int i = lane; i < RPW * HH; i += 32) {
    int r = i >> 6, c = i & 63;
    hbuf[lrow + r][c] = 0.f;
  }
  __syncthreads();

  // ---- per-lane elementwise constants (feature index m = t*16 + l16) ----
  float winc[4], binc[4], lnwc[4], lnbc[4], dec[4];
#pragma unroll
  for (int t = 0; t < 4; ++t) {
    int m = t * 16 + l16;
    winc[t] = w_in[m];
    binc[t] = b_in[m];
    lnwc[t] = ln_w[m];
    lnbc[t] = ln_b[m];
    dec[t]  = 1.f - DTC / softplusf(tau_p[m]);
  }

  // ---- W_rec B-fragments, register-resident for the whole recurrence ----
  // B[k][n] = W_rec[k][n]; lane holds (W_rec[kb][m], W_rec[kb+1][m]), kb = 4c + 2*hi.
  v2f bw[4][16];
#pragma unroll
  for (int c = 0; c < 16; ++c) {
    int kb = 4 * c + 2 * hi;
#pragma unroll
    for (int t = 0; t < 4; ++t) {
      int m = t * 16 + l16;
      v2f b;
      b.x = W_rec[kb * HH + m];
      b.y = W_rec[(kb + 1) * HH + m];
      bw[t][c] = b;
    }
  }

  // h also kept in registers in C/D layout: hreg[t][r] = h[row = lrow+8*hi+r][t*16+l16]
  float hreg[4][8];
#pragma unroll
  for (int t = 0; t < 4; ++t)
#pragma unroll
    for (int r = 0; r < 8; ++r) hreg[t][r] = 0.f;

  const v8f vzero = {0.f, 0.f, 0.f, 0.f, 0.f, 0.f, 0.f, 0.f};

  // =========================== Phase 1: recurrence ===========================
  for (int step = 0; step < WIN; ++step) {
    v8f acc[4];
#pragma unroll
    for (int t = 0; t < 4; ++t) acc[t] = vzero;

    // pre = h @ W_rec : 16 K-chunks x 4 N-tiles of V_WMMA_F32_16X16X4_F32
#pragma unroll
    for (int c = 0; c < 16; ++c) {
      int kb = 4 * c + 2 * hi;
      v2f a = *(const v2f*)&hbuf[lrow + l16][kb];   // A: h[row=l16][kb..kb+1]
#pragma unroll
      for (int t = 0; t < 4; ++t) acc[t] = WMMA_F32(a, bw[t][c], acc[t]);
    }

    // scalar input x_t per batch row (broadcast reads from LDS)
    float xt[8];
#pragma unroll
    for (int r = 0; r < 8; ++r) xt[r] = yT[step][lrow + 8 * hi + r];

    // elementwise: +x_t*w_in+b_in, layernorm over H, tanh, leaky update, clip
#pragma unroll
    for (int r = 0; r < 8; ++r) {
      float pre[4];
      float s = 0.f, q = 0.f;
#pragma unroll
      for (int t = 0; t < 4; ++t) {
        float p = acc[t][r] + xt[r] * winc[t] + binc[t];
        pre[t] = p;
        s += p;
        q += p * p;
      }
      // reduce across the 16 lanes of this half (row lives in one half under wave32)
#pragma unroll
      for (int msk = 1; msk <= 8; msk <<= 1) {
        s += __shfl_xor(s, msk, 32);
        q += __shfl_xor(q, msk, 32);
      }
      float mean = s * (1.f / 64.f);
      float var  = q * (1.f / 64.f) - mean * mean;
      float rstd = rsqrtf(var + LNEPS);
#pragma unroll
      for (int t = 0; t < 4; ++t) {
        float f  = fast_tanh((pre[t] - mean) * rstd * lnwc[t] + lnbc[t]);
        float hv = hreg[t][r] * dec[t] + DTC * f;
        hv = fminf(10.f, fmaxf(-10.f, hv));
        hreg[t][r] = hv;
        hbuf[lrow + 8 * hi + r][t * 16 + l16] = hv;   // D-layout -> row-major for next A-frags
      }
    }
  }

  // ======================= Phase 2: x = h @ Wx0^T + bx0 ======================
  {
    v8f acc[4];
#pragma unroll
    for (int t = 0; t < 4; ++t) acc[t] = vzero;
#pragma unroll
    for (int c = 0; c < 16; ++c) {
      int kb = 4 * c + 2 * hi;
      v2f a = *(const v2f*)&hbuf[lrow + l16][kb];
#pragma unroll
      for (int t = 0; t < 4; ++t) {
        v2f b = *(const v2f*)&Wx0[(t * 16 + l16) * HH + kb];  // B[k][a] = Wx0[a][k]
        acc[t] = WMMA_F32(a, b, acc[t]);
      }
    }
#pragma unroll
    for (int t = 0; t < 4; ++t) {
      float bx = bx0[t * 16 + l16];
#pragma unroll
      for (int r = 0; r < 8; ++r)
        hbuf[lrow + 8 * hi + r][t * 16 + l16] = acc[t][r] + bx;
    }
  }

  // ============================ Phase 3: ISTA (K=8) ==========================
  for (int k = 0; k < KIT; ++k) {
    const float* __restrict__ Wek = We + (size_t)k * AD * WIN;
    const float* __restrict__ Wrk = Wr + (size_t)k * AD * AD;

    v8f acc[4];
#pragma unroll
    for (int t = 0; t < 4; ++t) acc[t] = vzero;

    // z += y @ We[k]^T : 25 K-chunks over WIN=100
#pragma unroll
    for (int c = 0; c < 25; ++c) {
      int kt = 4 * c + 2 * hi;
      v2f a;
      a.x = yT[kt][lrow + l16];
      a.y = yT[kt + 1][lrow + l16];
#pragma unroll
      for (int t = 0; t < 4; ++t) {
        v2f b = *(const v2f*)&Wek[(size_t)(t * 16 + l16) * WIN + kt]; // B[t'][a]=We[k][a][t'] (L2-hot)
        acc[t] = WMMA_F32(a, b, acc[t]);
      }
    }
    // z += x @ Wr[k]^T : 16 K-chunks over A=64
#pragma unroll
    for (int c = 0; c < 16; ++c) {
      int kb = 4 * c + 2 * hi;
      v2f a = *(const v2f*)&hbuf[lrow + l16][kb];
#pragma unroll
      for (int t = 0; t < 4; ++t) {
        v2f b = *(const v2f*)&Wrk[(size_t)(t * 16 + l16) * AD + kb];
        acc[t] = WMMA_F32(a, b, acc[t]);
      }
    }
    // soft threshold, write x back (wave-private LDS, in-order DS keeps RAW safe)
#pragma unroll
    for (int t = 0; t < 4; ++t) {
      int aa = t * 16 + l16;
      float bek = be[k * AD + aa];
      float lam = softplusf(thr[k * AD + aa]);
#pragma unroll
      for (int r = 0; r < 8; ++r) {
        float z = acc[t][r] + bek;
        float m = fmaxf(fabsf(z) - lam, 0.f);
        hbuf[lrow + 8 * hi + r][aa] = copysignf(m, z);
      }
    }
  }

  // ===================== Phase 4: head + write outputs =======================
  __syncthreads();

  // x out: (B, A) flat after the power segment
  for (int i = tid; i < RPB * AD; i += blockDim.x) {
    int r = i >> 6, c = i & 63;
    out_x[grow0 * AD + (size_t)i] = hbuf[r][c];
  }
  // power out: (B, NAPP)
  for (int i = tid; i < RPB * NAPPL; i += blockDim.x) {
    int r = i >> 2, p = i & 3;
    float s = head_b[p];
#pragma unroll
    for (int a = 0; a < AD; ++a) s += hbuf[r][a] * head_W[p * AD + a];
    out_power[(grow0 + (size_t)r) * NAPPL + p] = s;
  }
}

extern "C" void kernel_launch(void* const* d_in, const int* in_sizes, int n_in,
                              void* d_out, int out_size, void* d_ws, size_t ws_size,
                              hipStream_t stream) {
  (void)in_sizes; (void)n_in; (void)d_ws; (void)ws_size; (void)out_size;
  const float* y      = (const float*)d_in[0];
  const float* w_in   = (const float*)d_in[1];
  const float* b_in   = (const float*)d_in[2];
  const float* tau_p  = (const float*)d_in[3];
  const float* W_rec  = (const float*)d_in[4];
  const float* ln_w   = (const float*)d_in[5];
  const float* ln_b   = (const float*)d_in[6];
  const float* Wx0    = (const float*)d_in[7];
  const float* bx0    = (const float*)d_in[8];
  const float* We     = (const float*)d_in[9];
  const float* be     = (const float*)d_in[10];
  const float* Wr     = (const float*)d_in[11];
  const float* thr    = (const float*)d_in[12];
  const float* head_W = (const float*)d_in[13];
  const float* head_b = (const float*)d_in[14];

  float* out_power = (float*)d_out;                        // (B, NAPP)
  float* out_x     = (float*)d_out + (size_t)NB * NAPPL;   // (B, A)

  dim3 grid(NB / RPB);   // 512 blocks
  dim3 block(128);       // 4 wave32
  hipLaunchKernelGGL(pinn_lista_fused, grid, block, 0, stream,
                     y, w_in, b_in, tau_p, W_rec, ln_w, ln_b, Wx0, bx0,
                     We, be, Wr, thr, head_W, head_b, out_power, out_x);
}